// VAE_3444563771689
// MI455X (gfx1250) — compile-verified
//
#include <hip/hip_runtime.h>

#define NN 50000
#define NE 800000
#define KC 32          // K-chunk staged in LDS per iteration
#define SA_STRIDE 36   // 32 + pad, multiple of 4 (16B-aligned float4 stores, spreads banks)
#define SB_STRIDE 68   // 64 + pad, multiple of 4

typedef __attribute__((ext_vector_type(2))) float v2f;
typedef __attribute__((ext_vector_type(8))) float v8f;

// ---------------- utility ----------------
__global__ void zero_f32(float* __restrict__ p, int n) {
  int i = blockIdx.x * blockDim.x + threadIdx.x;
  int stride = gridDim.x * blockDim.x;
  for (; i < n; i += stride) p[i] = 0.f;
}

__global__ void deg_count(const int* __restrict__ dst, float* __restrict__ deg, int E) {
  int e = blockIdx.x * blockDim.x + threadIdx.x;
  if (e < E) atomicAdd(&deg[dst[e]], 1.f);
}

__global__ void rdeg_inv(float* __restrict__ deg, int n) {
  int i = blockIdx.x * blockDim.x + threadIdx.x;
  if (i < n) deg[i] = 1.f / fmaxf(deg[i], 1.f);
}

// ------------- edge scatter-add (mean aggregation numerator) -------------
__global__ void scatter_add(const float* __restrict__ h, const int* __restrict__ src,
                            const int* __restrict__ dst, float* __restrict__ msg,
                            int E, int F, int log2chunks) {
  int tid = blockIdx.x * blockDim.x + threadIdx.x;
  int total = E << log2chunks;
  if (tid >= total) return;
  int e = tid >> log2chunks;
  int c = (tid & ((1 << log2chunks) - 1)) << 2;
  int s = src[e], d = dst[e];
  float4 v = *(const float4*)(h + (size_t)s * F + c);
  float* p = msg + (size_t)d * F + c;
  atomicAdd(p + 0, v.x);
  atomicAdd(p + 1, v.y);
  atomicAdd(p + 2, v.z);
  atomicAdd(p + 3, v.w);
}

// ------------- fused SAGE GEMM: out = relu?((msg*rdeg)@Wl + h@Wr + b) -------------
// block = 128 threads (4 waves) -> 16(M) x 64(N) output tile.
// Per K-chunk of 32: coalesced float4 staging of msg/h rows and Wl/Wr rows into LDS,
// then each wave runs f32 WMMA 16x16x4 on its own 16-col subtile (4x LDS reuse of A).
__global__ void sage_gemm_wmma(const float* __restrict__ msg, const float* __restrict__ rdeg,
                               const float* __restrict__ h,
                               const float* __restrict__ Wl, const float* __restrict__ Wr,
                               const float* __restrict__ bias, float* __restrict__ out,
                               int K, int F, int nblocks_n, int do_relu) {
  __shared__ float sA[2][16 * SA_STRIDE];   // [0]=msg chunk, [1]=h chunk (16 rows x 32 k)
  __shared__ float sB[2][KC * SB_STRIDE];   // [0]=Wl chunk,  [1]=Wr chunk (32 k x 64 n)

  int t = threadIdx.x;
  int wave = t >> 5, lane = t & 31;
  int mBase = (blockIdx.x / nblocks_n) * 16;
  int nBase = (blockIdx.x % nblocks_n) * 64;

  // WMMA f32 16x16x4 wave32 layouts (ISA tables):
  // A: lanes 0-15 -> M=lane, K={0,1}; lanes 16-31 -> M=lane-16, K={2,3}
  // B: symmetric with N in place of M
  // C: VGPR v -> M = v + 8*(lane>=16), N = lane&15
  int m16  = lane & 15;
  int kb   = (lane >> 4) << 1;
  int nloc = wave * 16 + (lane & 15);       // this wave's column within the 64-wide tile
  int ncol = nBase + nloc;

  float scale = rdeg[mBase + m16];
  v8f acc = {0.f, 0.f, 0.f, 0.f, 0.f, 0.f, 0.f, 0.f};

  // staging index precompute (fully coalesced float4 loads)
  int ar = t >> 3, ac = (t & 7) << 2;       // A: 16 rows x 32 cols, 1 float4/thread
  int br = t >> 4, bc = (t & 15) << 2;      // B: 8 rows x 64 cols per pass, 4 passes

  for (int kc = 0; kc < K; kc += KC) {
    // ---- stage chunk into LDS ----
    {
      float4 va0 = *(const float4*)(msg + (size_t)(mBase + ar) * K + kc + ac);
      float4 va1 = *(const float4*)(h   + (size_t)(mBase + ar) * K + kc + ac);
      *(float4*)(&sA[0][ar * SA_STRIDE + ac]) = va0;
      *(float4*)(&sA[1][ar * SA_STRIDE + ac]) = va1;
#pragma unroll
      for (int pass = 0; pass < 4; pass++) {
        int row = br + pass * 8;
        float4 vb0 = *(const float4*)(Wl + (size_t)(kc + row) * F + nBase + bc);
        float4 vb1 = *(const float4*)(Wr + (size_t)(kc + row) * F + nBase + bc);
        *(float4*)(&sB[0][row * SB_STRIDE + bc]) = vb0;
        *(float4*)(&sB[1][row * SB_STRIDE + bc]) = vb1;
      }
    }
    // prefetch next chunk while we compute this one (global_prefetch_b8)
    if (kc + KC < K) {
      __builtin_prefetch(msg + (size_t)(mBase + ar) * K + kc + KC + ac, 0, 0);
      __builtin_prefetch(h   + (size_t)(mBase + ar) * K + kc + KC + ac, 0, 0);
      __builtin_prefetch(Wl + (size_t)(kc + KC + br) * F + nBase + bc, 0, 0);
      __builtin_prefetch(Wr + (size_t)(kc + KC + br) * F + nBase + bc, 0, 0);
    }
    __syncthreads();

    // ---- compute: 8 k-steps x 2 operand pairs ----
#pragma unroll
    for (int k = 0; k < KC; k += 4) {
      v2f a, b;
      // part 1: mean-aggregated neighbors @ Wl (rdeg folded into A)
      a.x = sA[0][m16 * SA_STRIDE + k + kb] * scale;
      a.y = sA[0][m16 * SA_STRIDE + k + kb + 1] * scale;
      b.x = sB[0][(k + kb) * SB_STRIDE + nloc];
      b.y = sB[0][(k + kb + 1) * SB_STRIDE + nloc];
      acc = __builtin_amdgcn_wmma_f32_16x16x4_f32(false, a, false, b, (short)0, acc,
                                                  false, false);
      // part 2: self features @ Wr
      a.x = sA[1][m16 * SA_STRIDE + k + kb];
      a.y = sA[1][m16 * SA_STRIDE + k + kb + 1];
      b.x = sB[1][(k + kb) * SB_STRIDE + nloc];
      b.y = sB[1][(k + kb + 1) * SB_STRIDE + nloc];
      acc = __builtin_amdgcn_wmma_f32_16x16x4_f32(false, a, false, b, (short)0, acc,
                                                  false, false);
    }
    __syncthreads();
  }

  float bv = bias[ncol];
  int mhalf = mBase + ((lane >> 4) << 3);
#pragma unroll
  for (int v = 0; v < 8; v++) {
    float val = acc[v] + bv;
    if (do_relu) val = fmaxf(val, 0.f);
    out[(size_t)(mhalf + v) * F + ncol] = val;
  }
}

// ------------- batch norm: column stats then apply -------------
__global__ void bn_stats(const float* __restrict__ h, float* __restrict__ st, int M, int F) {
  int c = threadIdx.x;  // blockDim.x == F
  float s = 0.f, s2 = 0.f;
  for (int r = blockIdx.x; r < M; r += gridDim.x) {
    float v = h[(size_t)r * F + c];
    s += v;
    s2 += v * v;
  }
  atomicAdd(&st[c], s);
  atomicAdd(&st[F + c], s2);
}

__global__ void bn_apply(float* __restrict__ h, const float* __restrict__ st,
                         const float* __restrict__ g, const float* __restrict__ be,
                         int M, int F) {
  int i = blockIdx.x * blockDim.x + threadIdx.x;
  if (i >= M * F) return;
  int c = i & (F - 1);
  float inv = 1.f / (float)M;
  float mean = st[c] * inv;
  float var = st[F + c] * inv - mean * mean;
  h[i] = g[c] * (h[i] - mean) * rsqrtf(var + 1e-5f) + be[c];
}

// ------------- z = mean + exp(log_std) * eps -------------
__global__ void reparam(const float* __restrict__ h3, const float* __restrict__ eps,
                        float* __restrict__ z, int M) {
  int i = blockIdx.x * blockDim.x + threadIdx.x;
  if (i >= M * 64) return;
  int r = i >> 6, c = i & 63;
  const float* row = h3 + (size_t)r * 128;
  z[i] = row[c] + expf(row[64 + c]) * eps[i];
}

extern "C" void kernel_launch(void* const* d_in, const int* in_sizes, int n_in,
                              void* d_out, int out_size, void* d_ws, size_t ws_size,
                              hipStream_t stream) {
  const float* x   = (const float*)d_in[0];
  const int*   src = (const int*)d_in[1];
  const int*   dst = (const int*)d_in[2];
  const float* eps = (const float*)d_in[3];
  const float* W1l = (const float*)d_in[4];
  const float* W1r = (const float*)d_in[5];
  const float* b1  = (const float*)d_in[6];
  const float* g1  = (const float*)d_in[7];
  const float* be1 = (const float*)d_in[8];
  const float* W2l = (const float*)d_in[9];
  const float* W2r = (const float*)d_in[10];
  const float* b2  = (const float*)d_in[11];
  const float* g2  = (const float*)d_in[12];
  const float* be2 = (const float*)d_in[13];
  const float* W3l = (const float*)d_in[14];
  const float* W3r = (const float*)d_in[15];
  const float* b3  = (const float*)d_in[16];
  float* zout = (float*)d_out;
  (void)in_sizes; (void)n_in; (void)out_size; (void)ws_size;

  char* ws = (char*)d_ws;
  size_t off = 0;
  auto carve = [&](size_t bytes) -> float* {
    char* p = ws + off;
    off = (off + bytes + 255) & ~(size_t)255;
    return (float*)p;
  };
  float* rdeg = carve((size_t)NN * 4);            // degrees -> reciprocal degrees
  float* msg  = carve((size_t)NN * 256 * 4);      // aggregation numerator (max F=256)
  float* h1   = carve((size_t)NN * 128 * 4);      // layer-1 output, reused for layer-3 output
  float* h2   = carve((size_t)NN * 256 * 4);      // layer-2 output
  float* st   = carve(512 * 4);                   // BN sums / sumsq

  const int M = NN, E = NE;

  // degree / reciprocal degree (shared by all layers)
  zero_f32<<<256, 256, 0, stream>>>(rdeg, M);
  deg_count<<<(E + 255) / 256, 256, 0, stream>>>(dst, rdeg, E);
  rdeg_inv<<<(M + 255) / 256, 256, 0, stream>>>(rdeg, M);

  // ---- layer 1: 128 -> 128, relu + BN ----
  zero_f32<<<2048, 256, 0, stream>>>(msg, M * 128);
  scatter_add<<<(E * 32 + 255) / 256, 256, 0, stream>>>(x, src, dst, msg, E, 128, 5);
  sage_gemm_wmma<<<3125 * 2, 128, 0, stream>>>(msg, rdeg, x, W1l, W1r, b1, h1,
                                               128, 128, 2, 1);
  zero_f32<<<1, 256, 0, stream>>>(st, 256);
  bn_stats<<<512, 128, 0, stream>>>(h1, st, M, 128);
  bn_apply<<<(M * 128 + 255) / 256, 256, 0, stream>>>(h1, st, g1, be1, M, 128);

  // ---- layer 2: 128 -> 256, relu + BN ----
  zero_f32<<<2048, 256, 0, stream>>>(msg, M * 128);
  scatter_add<<<(E * 32 + 255) / 256, 256, 0, stream>>>(h1, src, dst, msg, E, 128, 5);
  sage_gemm_wmma<<<3125 * 4, 128, 0, stream>>>(msg, rdeg, h1, W2l, W2r, b2, h2,
                                               128, 256, 4, 1);
  zero_f32<<<2, 256, 0, stream>>>(st, 512);
  bn_stats<<<512, 256, 0, stream>>>(h2, st, M, 256);
  bn_apply<<<(M * 256 + 255) / 256, 256, 0, stream>>>(h2, st, g2, be2, M, 256);

  // ---- layer 3: 256 -> 128, linear (mean | log_std) ----
  zero_f32<<<4096, 256, 0, stream>>>(msg, M * 256);
  scatter_add<<<(E * 64 + 255) / 256, 256, 0, stream>>>(h2, src, dst, msg, E, 256, 6);
  sage_gemm_wmma<<<3125 * 2, 128, 0, stream>>>(msg, rdeg, h2, W3l, W3r, b3, h1,
                                               256, 128, 2, 0);

  // ---- reparameterize ----
  reparam<<<(M * 64 + 255) / 256, 256, 0, stream>>>(h1, eps, zout, M);
}